// DGMF_9990093931296
// MI455X (gfx1250) — compile-verified
//
#include <hip/hip_runtime.h>
#include <math.h>

// Problem constants (match reference)
constexpr int Vn = 2;
constexpr int Nn = 100000;
constexpr int Kn = 8;
constexpr int Dn = 128;
constexpr int NVout = Nn * Vn;          // 200000
constexpr int NTILES = Nn / 16;          // 6250 (exact)
constexpr float LOG_2PI_F = 1.8378770664093453f;

// Workspace layout (float offsets)
constexpr size_t WS_INVB = 0;                       // V*K*D*D = 262144 (B-fragment order)
constexpr size_t WS_B    = WS_INVB + (size_t)Vn*Kn*Dn*Dn;   // V*K*D = 2048
constexpr size_t WS_KOFF = WS_B + (size_t)Vn*Kn*Dn;         // 16
constexpr size_t WS_PEN  = WS_KOFF + Vn*Kn;                 // 16
constexpr size_t WS_TILE = WS_PEN + Vn*Kn;                  // 6250

typedef float v2f __attribute__((ext_vector_type(2)));
typedef float v8f __attribute__((ext_vector_type(8)));

// ---------------------------------------------------------------------------
// Kernel 1: per (v,k): invert sigma+1e-6I (in-place Gauss-Jordan, no pivoting;
// sigma = I + PSD perturbation is diagonally dominant), logdet from pivots,
// b = inv*mu, fold all per-(v,k) constants into koff. Export inverse in WMMA
// B-fragment order for coalesced b64 loads in the hot kernel.
// ---------------------------------------------------------------------------
__global__ __launch_bounds__(128) void prep_kernel(const float* __restrict__ phi,
                                                   const float* __restrict__ mu,
                                                   const float* __restrict__ sigma,
                                                   float* __restrict__ ws)
{
    __shared__ float S[Dn * Dn];   // exactly 64 KB
    const int vk  = blockIdx.x;            // 0..15
    const int v   = vk / Kn;
    const int k   = vk % Kn;
    const int tid = threadIdx.x;           // 0..127, owns column `tid`
    const float* sig = sigma + (size_t)vk * Dn * Dn;

    // Load + stabilize diagonal
    for (int i = tid; i < Dn * Dn; i += 128) {
        float x = sig[i];
        if ((i % (Dn + 1)) == 0) x += 1e-6f;
        S[i] = x;
    }
    __syncthreads();

    float logdet = 0.0f;   // accumulated redundantly by every thread
    for (int p = 0; p < Dn; ++p) {
        const float piv = S[p * Dn + p];
        const float d   = 1.0f / piv;
        logdet += logf(piv);
        __syncthreads();                       // everyone read piv before row-p write
        // Scale pivot row: thread tid owns entry (p, tid)
        S[p * Dn + tid] = (tid == p) ? d : S[p * Dn + tid] * d;
        __syncthreads();
        // Eliminate: thread tid updates its own column (reads untouched column p
        // and the scaled pivot row; no cross-thread write hazards)
        if (tid != p) {
            const float rpj = S[p * Dn + tid];
            for (int i = 0; i < Dn; ++i) {
                if (i == p) continue;
                S[i * Dn + tid] -= S[i * Dn + p] * rpj;
            }
        }
        __syncthreads();
        // Column p update: thread tid handles row tid's entry of column p
        if (tid != p) S[tid * Dn + p] *= -d;
        __syncthreads();
    }
    // S now holds inv(sigma_stable).

    // Export in WMMA B-fragment order: o = ((j*32 + f)*32 + lane)*2 + r
    // value = inv[4f + 2*(lane>=16) + r][16j + lane%16]
    float* invB = ws + WS_INVB + (size_t)vk * Dn * Dn;
    for (int o = tid; o < Dn * Dn; o += 128) {
        const int r   = o & 1;
        const int l   = (o >> 1) & 31;
        const int f   = (o >> 6) & 31;
        const int j   = o >> 11;
        const int c   = 4 * f + 2 * (l >> 4) + r;
        const int col = 16 * j + (l & 15);
        invB[o] = S[c * Dn + col];
    }

    // b = inv * mu  (thread tid computes row tid)
    const float* muvk = mu + (size_t)vk * Dn;
    float bt = 0.0f;
    for (int jx = 0; jx < Dn; ++jx) bt += S[tid * Dn + jx] * muvk[jx];
    ws[WS_B + (size_t)vk * Dn + tid] = bt;
    __syncthreads();          // all reads of S done
    S[tid] = bt;              // stash b in (now reusable) row 0
    __syncthreads();

    if (tid == 0) {
        float c = 0.0f;
        for (int t = 0; t < Dn; ++t) c += muvk[t] * S[t];       // mu^T inv mu
        // log_softmax(phi[v])
        float mxp = -INFINITY;
        for (int kk = 0; kk < Kn; ++kk) mxp = fmaxf(mxp, phi[v * Kn + kk]);
        float sp = 0.0f;
        for (int kk = 0; kk < Kn; ++kk) sp += expf(phi[v * Kn + kk] - mxp);
        const float logpi = phi[v * Kn + k] - (mxp + logf(sp));
        const float ld = fmaxf(logdet, logf(1e-6f));            // det clip
        ws[WS_KOFF + vk] = -0.5f * c - 0.5f * (float)Dn * LOG_2PI_F - 0.5f * ld + logpi;
        // penalty contribution: sum 1/(diag(sigma_stable)+1e-12)
        float pen = 0.0f;
        for (int dd = 0; dd < Dn; ++dd)
            pen += 1.0f / (sig[dd * (Dn + 1)] + 1e-6f + 1e-12f);
        ws[WS_PEN + vk] = pen;
    }
}

// ---------------------------------------------------------------------------
// Kernel 2: one wave per 16-row tile. G = Z_tile * inv_sigma via f32 WMMA,
// maha diag via elementwise C-layout multiply + half-wave shuffle reduce,
// logsumexp over K, softmax weights over views.
// ---------------------------------------------------------------------------
__global__ __launch_bounds__(256) void energy_kernel(const float* __restrict__ z,
                                                     const float* __restrict__ ws,
                                                     float* __restrict__ out,
                                                     float* __restrict__ wsTile)
{
    __shared__ float lp_lds[8 * Kn * 16];   // [wave][k][row]
    const int wid  = threadIdx.x >> 5;
    const int lane = threadIdx.x & 31;
    const int tile = blockIdx.x * 8 + wid;
    if (tile >= NTILES) return;             // wave-uniform: EXEC stays all-1 for WMMA
    const int n0 = tile * 16;
    const int hl = lane >> 4;               // half-wave id
    const int ln = lane & 15;

    float e[Vn];
    for (int v = 0; v < Vn; ++v) {
        const float* zv = z + (size_t)v * Nn * Dn;

        // Z tile in A-fragment layout: lane ln = row, K chunk 4f + 2*hl + {0,1}
        v2f za[32];
        const float* zrow = zv + (size_t)(n0 + ln) * Dn + 2 * hl;
        #pragma unroll
        for (int f = 0; f < 32; ++f) za[f] = *(const v2f*)(zrow + 4 * f);

        for (int k = 0; k < Kn; ++k) {
            const int vk = v * Kn + k;
            float acc[8];
            #pragma unroll
            for (int r = 0; r < 8; ++r) acc[r] = 0.0f;
            const float* bvec = ws + WS_B + (size_t)vk * Dn;

            for (int j = 0; j < 8; ++j) {   // output column tiles
                const float* bf = ws + WS_INVB + ((size_t)(vk * 8 + j) * 32) * 64 + 2 * lane;
                v8f C = {0.f, 0.f, 0.f, 0.f, 0.f, 0.f, 0.f, 0.f};
                #pragma unroll
                for (int f = 0; f < 32; ++f) {
                    v2f bb = *(const v2f*)(bf + f * 64);   // coalesced b64
                    C = __builtin_amdgcn_wmma_f32_16x16x4_f32(
                            false, za[f], false, bb, (short)0, C, false, false);
                }
                // diag extraction: C[r] holds G[row=r+8*hl, col=16j+ln]
                const float b2 = 2.0f * bvec[j * 16 + ln];
                #pragma unroll
                for (int r = 0; r < 8; ++r) {
                    const float zc = zv[(size_t)(n0 + r + 8 * hl) * Dn + j * 16 + ln];
                    acc[r] += (C[r] - b2) * zc;
                }
            }
            // reduce over columns: sum across each 16-lane half
            #pragma unroll
            for (int r = 0; r < 8; ++r) {
                float a = acc[r];
                a += __shfl_xor(a, 1, 16);
                a += __shfl_xor(a, 2, 16);
                a += __shfl_xor(a, 4, 16);
                a += __shfl_xor(a, 8, 16);
                acc[r] = a;
            }
            const float koff = ws[WS_KOFF + vk];
            if (ln == 0) {  // lane 0 -> rows 0..7, lane 16 -> rows 8..15
                #pragma unroll
                for (int r = 0; r < 8; ++r)
                    lp_lds[wid * 128 + k * 16 + hl * 8 + r] = -0.5f * acc[r] + koff;
            }
        }
        // logsumexp over K for row = ln (same-wave LDS, lockstep ordering)
        float lp[Kn];
        float mx = -INFINITY;
        #pragma unroll
        for (int k = 0; k < Kn; ++k) {
            lp[k] = lp_lds[wid * 128 + k * 16 + ln];
            mx = fmaxf(mx, lp[k]);
        }
        float s = 0.0f;
        #pragma unroll
        for (int k = 0; k < Kn; ++k) s += expf(lp[k] - mx);
        e[v] = -(mx + logf(s));
    }

    const float e0 = e[0], e1 = e[1];
    float tsum = e0 + e1;                // per-row total, reduce over 16 rows
    tsum += __shfl_xor(tsum, 1, 16);
    tsum += __shfl_xor(tsum, 2, 16);
    tsum += __shfl_xor(tsum, 4, 16);
    tsum += __shfl_xor(tsum, 8, 16);

    if (lane < 16) {
        const int n = n0 + lane;
        out[2 * n]     = e0;
        out[2 * n + 1] = e1;
        const float m2 = fmaxf(-e0, -e1);
        const float w0 = expf(-e0 - m2);
        const float w1 = expf(-e1 - m2);
        const float si = 1.0f / (w0 + w1);
        out[NVout + 2 * n]     = w0 * si;
        out[NVout + 2 * n + 1] = w1 * si;
    }
    if (lane == 0) wsTile[tile] = tsum;
}

// ---------------------------------------------------------------------------
// Kernel 3: deterministic reduction of per-tile energy sums and penalties.
// ---------------------------------------------------------------------------
__global__ __launch_bounds__(256) void finalize_kernel(const float* __restrict__ ws,
                                                       float* __restrict__ out)
{
    __shared__ float red[256];
    const int tid = threadIdx.x;
    float s = 0.0f;
    for (int i = tid; i < NTILES; i += 256) s += ws[WS_TILE + i];
    red[tid] = s;
    __syncthreads();
    for (int step = 128; step > 0; step >>= 1) {
        if (tid < step) red[tid] += red[tid + step];
        __syncthreads();
    }
    if (tid == 0) {
        out[2 * NVout] = red[0];
        float pen = 0.0f;
        for (int i = 0; i < Vn * Kn; ++i) pen += ws[WS_PEN + i];
        out[2 * NVout + 1] = pen;
    }
}

// ---------------------------------------------------------------------------
extern "C" void kernel_launch(void* const* d_in, const int* in_sizes, int n_in,
                              void* d_out, int out_size, void* d_ws, size_t ws_size,
                              hipStream_t stream)
{
    const float* z     = (const float*)d_in[0];   // [V,N,D]
    const float* phi   = (const float*)d_in[1];   // [V,K]
    const float* mu    = (const float*)d_in[2];   // [V,K,D]
    const float* sigma = (const float*)d_in[3];   // [V,K,D,D]
    float* out = (float*)d_out;
    float* ws  = (float*)d_ws;

    prep_kernel<<<Vn * Kn, 128, 0, stream>>>(phi, mu, sigma, ws);
    energy_kernel<<<(NTILES + 7) / 8, 256, 0, stream>>>(z, ws, out, ws + WS_TILE);
    finalize_kernel<<<1, 256, 0, stream>>>(ws, out);
}